// MultiHeadSelfAttention1_50182397887198
// MI455X (gfx1250) — compile-verified
//
#include <hip/hip_runtime.h>
#include <math.h>

// MI455X / gfx1250: wave32, WMMA f32 16x16x4, 320 KB LDS per WGP
typedef __attribute__((ext_vector_type(2))) float v2f;
typedef __attribute__((ext_vector_type(8))) float v8f;

#define BATCH 4
#define SEQ   4096
#define DIM   128
#define BS    (BATCH * SEQ)     // 16384 total rows
#define NQKV  (3 * DIM)         // 384

#define SC_STRIDE   4100        // 4096 + 4: stride%8==4 -> lanes16-31 bank offset 32
#define AT_STRIDE   132         // 128 + 4 : same trick for the attn tile

static __device__ __forceinline__ v8f wmma4(v2f a, v2f b, v8f c) {
    // D = A(16x4 f32) x B(4x16 f32) + C(16x16 f32)
    return __builtin_amdgcn_wmma_f32_16x16x4_f32(
        /*neg_a=*/false, a, /*neg_b=*/false, b,
        /*c_mod=*/(short)0, c, /*reuse_a=*/false, /*reuse_b=*/false);
}

static __device__ __forceinline__ v8f vzero8() {
    v8f z = {0.f, 0.f, 0.f, 0.f, 0.f, 0.f, 0.f, 0.f};
    return z;
}

// ---------------------------------------------------------------------------
// Kernel 1: qkv = x @ Wqkv^T + bqkv   (M=16384, N=384, K=128)
// One wave -> one 16x16 output tile. Scatter into separate q/k/v buffers.
// ---------------------------------------------------------------------------
__global__ void qkv_proj_kernel(const float* __restrict__ x,
                                const float* __restrict__ Wqkv,
                                const float* __restrict__ bqkv,
                                float* __restrict__ q,
                                float* __restrict__ k,
                                float* __restrict__ v) {
    const int wid  = blockIdx.x * (blockDim.x >> 5) + (threadIdx.x >> 5);
    const int lane = threadIdx.x & 31;
    const int mt = wid / (NQKV / 16);   // 0..1023
    const int nt = wid % (NQKV / 16);   // 0..23
    const int lrow = lane & 15;
    const int kh   = (lane >> 4) << 1;  // 0 for lanes 0-15, 2 for lanes 16-31

    const float* Arow = x    + (size_t)(mt * 16 + lrow) * DIM + kh;
    const float* Brow = Wqkv + (size_t)(nt * 16 + lrow) * DIM + kh;

    v8f acc = vzero8();
#pragma unroll
    for (int kk = 0; kk < DIM; kk += 4) {
        v2f a = *(const v2f*)(Arow + kk);
        v2f b = *(const v2f*)(Brow + kk);
        acc = wmma4(a, b, acc);
    }

    const int ncol = nt * 16 + lrow;
    const float bias = bqkv[ncol];
    float* dst; int nl;
    if (ncol < DIM)          { dst = q; nl = ncol; }
    else if (ncol < 2 * DIM) { dst = k; nl = ncol - DIM; }
    else                     { dst = v; nl = ncol - 2 * DIM; }

    const int mbase = mt * 16 + ((lane >> 4) << 3);
#pragma unroll
    for (int r = 0; r < 8; ++r)
        dst[(size_t)(mbase + r) * DIM + nl] = acc[r] + bias;
}

// ---------------------------------------------------------------------------
// Kernel 2: fully fused attention for one 16-row block.
//   Phase 1: masked scores Q(16x128) @ K^T -> LDS [16][4096] (8 waves x 512 cols)
//   Phase 2: row softmax in LDS; write normalized weights to HBM exactly once;
//            keep exp() values in LDS, fold 1/rowsum into phase 3.
//   Phase 3: attn(16x128) = exp-weights(LDS) @ V(global), scaled by 1/rowsum,
//            result into LDS attn tile.
//   Phase 4: out(16x128) = attn(LDS) @ Wo^T + bo -> HBM.
// Grid: 1024 blocks (4 batches x 256 row-tiles) x 256 threads (8 waves).
// Dynamic LDS: 272,000 B (fits CDNA5's 320 KB/WGP; one block per WGP).
// ---------------------------------------------------------------------------
__global__ void fused_attention_kernel(const float* __restrict__ q,
                                       const float* __restrict__ kmat,
                                       const float* __restrict__ v,
                                       const float* __restrict__ Wo,
                                       const float* __restrict__ bo,
                                       float* __restrict__ wts,
                                       float* __restrict__ out) {
    extern __shared__ float smem[];
    float* sc      = smem;                       // [16][SC_STRIDE] score rows
    float* red     = smem + 16 * SC_STRIDE;      // [256] partial reductions
    float* rowstat = red + 256;                  // [0..15]=rowmax, [16..31]=1/rowsum
    float* sattn   = rowstat + 32;               // [16][AT_STRIDE] attn tile

    const int tid  = threadIdx.x;
    const int wave = tid >> 5;                   // 0..7
    const int lane = tid & 31;
    const int b    = blockIdx.x >> 8;            // batch
    const int mt   = blockIdx.x & 255;           // 16-row tile within batch
    const int lrow = lane & 15;
    const int kh   = (lane >> 4) << 1;           // K-pair select: 0 or 2
    const int off8 = (lane >> 4) << 3;           // C row offset: 0 or 8

    const float* Q = q    + (size_t)b * SEQ * DIM;
    const float* K = kmat + (size_t)b * SEQ * DIM;
    const float* V = v    + (size_t)b * SEQ * DIM;
    const float NEG_INF = -__builtin_inff();

    // ---- Phase 1: masked scores into LDS --------------------------------
    const float* Arow = Q + (size_t)(mt * 16 + lrow) * DIM + kh;
#pragma unroll 1
    for (int s = 0; s < 4; ++s) {
        const int ns = (wave << 2) + s;          // 128-col strip id, 0..31
        const float* Bbase = K + (size_t)(ns * 128 + lrow) * DIM + kh;

        v8f acc[8];
#pragma unroll
        for (int t = 0; t < 8; ++t) acc[t] = vzero8();

#pragma unroll 2
        for (int kk = 0; kk < DIM; kk += 4) {
            v2f a = *(const v2f*)(Arow + kk);
#pragma unroll
            for (int t = 0; t < 8; ++t) {
                v2f bb = *(const v2f*)(Bbase + (size_t)t * 16 * DIM + kk);
                acc[t] = wmma4(a, bb, acc[t]);
            }
        }

        const int rowg_base = mt * 16;
#pragma unroll
        for (int t = 0; t < 8; ++t) {
            const int col = ns * 128 + t * 16 + lrow;
#pragma unroll
            for (int r = 0; r < 8; ++r) {
                const int rowl = off8 + r;
                const int d = (rowg_base + rowl) - col;
                const float val = (d >= -1 && d <= 1) ? NEG_INF : acc[t][r];
                sc[rowl * SC_STRIDE + col] = val;
            }
        }
    }
    __syncthreads();

    // ---- Phase 2: softmax over 16 rows of 4096 in LDS -------------------
    const int g = tid >> 4;                      // column group 0..15
    const int r = tid & 15;                      // row 0..15
    float* rowp = sc + r * SC_STRIDE + g;

    float m = NEG_INF;
#pragma unroll 4
    for (int i = 0; i < 256; ++i) m = fmaxf(m, rowp[i * 16]);
    red[r * 16 + g] = m;
    __syncthreads();
    if (tid < 16) {
        float mm = red[tid * 16];
#pragma unroll
        for (int j = 1; j < 16; ++j) mm = fmaxf(mm, red[tid * 16 + j]);
        rowstat[tid] = mm;
    }
    __syncthreads();
    m = rowstat[r];

    float ssum = 0.f;
#pragma unroll 4
    for (int i = 0; i < 256; ++i) {
        const float e = __expf(rowp[i * 16] - m);  // exp(-inf - m) == 0 -> mask
        rowp[i * 16] = e;
        ssum += e;
    }
    red[r * 16 + g] = ssum;
    __syncthreads();
    if (tid < 16) {
        float s = red[tid * 16];
#pragma unroll
        for (int j = 1; j < 16; ++j) s += red[tid * 16 + j];
        rowstat[16 + tid] = 1.f / s;
    }
    __syncthreads();

    // Single HBM write of the normalized weights block (coalesced).
    float* Wout = wts + (size_t)b * SEQ * SEQ + (size_t)(mt * 16) * SEQ;
    for (int idx = tid; idx < 16 * SEQ; idx += 256) {
        const int rr = idx >> 12;
        const int cc = idx & (SEQ - 1);
        Wout[(size_t)rr * SEQ + cc] = sc[rr * SC_STRIDE + cc] * rowstat[16 + rr];
    }

    // ---- Phase 3: attn = (exp-weights in LDS) @ V, scaled by 1/rowsum ---
    // Wave w owns output columns [w*16, w*16+16). 4 interleaved accumulators
    // break the 1024-step WMMA RAW chain.
    const int colw = (wave << 4) + lrow;
    const float* Vcol  = V + colw;
    const float* Aband = sc + lrow * SC_STRIDE + kh;

    v8f acc4[4];
#pragma unroll
    for (int j = 0; j < 4; ++j) acc4[j] = vzero8();

#pragma unroll 1
    for (int kk = 0; kk < SEQ; kk += 16) {
#pragma unroll
        for (int j = 0; j < 4; ++j) {
            const int kb = kk + (j << 2);
            v2f a;
            a[0] = Aband[kb];
            a[1] = Aband[kb + 1];
            const int kr = kb + kh;
            v2f bb;
            bb[0] = Vcol[(size_t)kr * DIM];
            bb[1] = Vcol[(size_t)(kr + 1) * DIM];
            acc4[j] = wmma4(a, bb, acc4[j]);
        }
    }
    v8f accf = (acc4[0] + acc4[1]) + (acc4[2] + acc4[3]);

#pragma unroll
    for (int rr = 0; rr < 8; ++rr) {
        const int rowl = off8 + rr;
        sattn[rowl * AT_STRIDE + colw] = accf[rr] * rowstat[16 + rowl];
    }
    __syncthreads();

    // ---- Phase 4: out = attn(LDS) @ Wo^T + bo ---------------------------
    const float* Brow = Wo + (size_t)((wave << 4) + lrow) * DIM + kh;
    const float* Aat  = sattn + lrow * AT_STRIDE + kh;

    v8f acc2[2];
    acc2[0] = vzero8();
    acc2[1] = vzero8();
#pragma unroll
    for (int kk = 0; kk < DIM; kk += 8) {
        v2f a0; a0[0] = Aat[kk];     a0[1] = Aat[kk + 1];
        v2f b0 = *(const v2f*)(Brow + kk);
        acc2[0] = wmma4(a0, b0, acc2[0]);
        v2f a1; a1[0] = Aat[kk + 4]; a1[1] = Aat[kk + 5];
        v2f b1 = *(const v2f*)(Brow + kk + 4);
        acc2[1] = wmma4(a1, b1, acc2[1]);
    }
    v8f acco = acc2[0] + acc2[1];

    const int col = (wave << 4) + lrow;
    const float bias = bo[col];
    float* Orow = out + (size_t)(b * SEQ + mt * 16) * DIM;
#pragma unroll
    for (int rr = 0; rr < 8; ++rr)
        Orow[(size_t)(off8 + rr) * DIM + col] = acco[rr] + bias;
}

// ---------------------------------------------------------------------------
extern "C" void kernel_launch(void* const* d_in, const int* in_sizes, int n_in,
                              void* d_out, int out_size, void* d_ws, size_t ws_size,
                              hipStream_t stream) {
    const float* x    = (const float*)d_in[0];   // [4,64,64,128]
    const float* Wqkv = (const float*)d_in[1];   // [384,128]
    const float* bqkv = (const float*)d_in[2];   // [384]
    const float* Wo   = (const float*)d_in[3];   // [128,128]
    const float* bo   = (const float*)d_in[4];   // [128]

    float* out = (float*)d_out;                  // [4,4096,128]    (output first)
    float* wts = out + (size_t)BS * DIM;         // [4,1,4096,4096] (weights second)

    float* q = (float*)d_ws;                     // [16384,128]
    float* k = q + (size_t)BS * DIM;
    float* v = k + (size_t)BS * DIM;             // total ws use: 24 MB

    // 1) QKV projection: 1024*24 = 24576 waves, 8 waves/block
    qkv_proj_kernel<<<3072, 256, 0, stream>>>(x, Wqkv, bqkv, q, k, v);

    // 2) Fused scores+softmax+WV+out-proj: 4*256 blocks x 8 waves,
    //    272,000 B dynamic LDS per block (<= 320 KB CDNA5 WGP LDS).
    const size_t lds_bytes =
        (size_t)(16 * SC_STRIDE + 256 + 32 + 16 * AT_STRIDE) * sizeof(float);
    fused_attention_kernel<<<BATCH * 256, 256, lds_bytes, stream>>>(
        q, k, v, Wo, bo, wts, out);
}